// GCN_55542517072486
// MI455X (gfx1250) — compile-verified
//
#include <hip/hip_runtime.h>
#include <hip/hip_bf16.h>

typedef __attribute__((ext_vector_type(2))) float v2f;
typedef __attribute__((ext_vector_type(8))) float v8f;

#define FD   64      // feature / hidden dim
#define GGR  512     // number of graphs
#define NCLS 10      // classes

// ---------------------------------------------------------------------------
// Prep kernels
// ---------------------------------------------------------------------------
__global__ void k_mask_x(const float* __restrict__ x, const unsigned char* __restrict__ mask,
                         float* __restrict__ h0, int N) {
    int g = blockIdx.x * blockDim.x + threadIdx.x;
    if (g >= N * FD) return;
    int n = g >> 6;
    h0[g] = mask[n] ? x[g] : 0.0f;
}

__global__ void k_deg_init(const unsigned char* __restrict__ mask, float* __restrict__ deg, int N) {
    int n = blockIdx.x * blockDim.x + threadIdx.x;
    if (n >= N) return;
    deg[n] = mask[n] ? 1.0f : 0.0f;
}

__global__ void k_edge_deg(const int* __restrict__ src, const int* __restrict__ dst,
                           const unsigned char* __restrict__ mask, float* __restrict__ deg, int E) {
    int e = blockIdx.x * blockDim.x + threadIdx.x;
    if (e >= E) return;
    int s = src[e], d = dst[e];
    if (mask[s] && mask[d]) atomicAdd(&deg[d], 1.0f);
}

__global__ void k_dinv_nself(float* __restrict__ deg_dinv, float* __restrict__ nself, int N) {
    int n = blockIdx.x * blockDim.x + threadIdx.x;
    if (n >= N) return;
    float d  = deg_dinv[n];
    float di = (d > 0.0f) ? rsqrtf(d) : 0.0f;
    deg_dinv[n] = di;
    nself[n]    = di * di;   // masked-out nodes have deg==0 -> di==0
}

__global__ void k_norm_e(const int* __restrict__ src, const int* __restrict__ dst,
                         const unsigned char* __restrict__ mask,
                         const float* __restrict__ dinv, float* __restrict__ ne, int E) {
    int e = blockIdx.x * blockDim.x + threadIdx.x;
    if (e >= E) return;
    int s = src[e], d = dst[e];
    ne[e] = (mask[s] && mask[d]) ? dinv[s] * dinv[d] : 0.0f;
}

// ---------------------------------------------------------------------------
// WMMA GEMM: out[M,64] = A[M,64] @ W[64,64]  (+bias, relu when BIAS_RELU)
// 256 threads = 8 waves; wave w computes rows [16*(8*bx+w), +16) x all 64 cols.
// V_WMMA_F32_16X16X4_F32, K=64 as 16 steps of 4. M must be a multiple of 16.
//
// W is staged into LDS *pre-swizzled into B-fragment order*:
//   Wl[kk][ct][lane][v]  (16*4*32*2 floats = 16 KB)
// where the B fragment element for (kk, ct, lane, v) is
//   W[4*kk + (lane>>4)*2 + v][ct*16 + (lane&15)]
// so each inner-loop B fragment is a single aligned ds_load_b64, no repacking,
// and the 32-lane b64 access spans all 64 LDS banks (conflict-free).
// ---------------------------------------------------------------------------
template <bool BIAS_RELU>
__global__ void k_gemm64(const float* __restrict__ A, const float* __restrict__ W,
                         const float* __restrict__ bias, float* __restrict__ out, int M) {
    __shared__ __align__(16) float Wl[16 * 4 * 32 * 2];
    int tid = threadIdx.x;
    for (int i = tid; i < 16 * 4 * 32 * 2; i += 256) {
        int v  = i & 1;
        int ln = (i >> 1) & 31;
        int ct = (i >> 6) & 3;
        int kk = i >> 8;
        int k  = 4 * kk + (ln >> 4) * 2 + v;
        int n  = ct * 16 + (ln & 15);
        Wl[i] = W[k * FD + n];
    }
    __syncthreads();

    int wave = tid >> 5;
    int lane = tid & 31;
    int r16  = lane & 15;   // row within A tile / col within 16-col group
    int hi   = lane >> 4;   // half-wave select

    int rowBase = (blockIdx.x * 8 + wave) * 16;
    if (rowBase >= M) return;   // wave-uniform -> EXEC stays all ones for WMMA

    v8f c0 = {}, c1 = {}, c2 = {}, c3 = {};
    const float* Arow  = A + (size_t)(rowBase + r16) * FD;
    const float* Bfrag = Wl + lane * 2;

#pragma unroll
    for (int kk = 0; kk < 16; ++kk) {
        int k0 = kk * 4 + hi * 2;                    // even -> 8B aligned
        v2f a  = *(const v2f*)(Arow + k0);           // A[r16][k0..k0+1]
        const float* Bk = Bfrag + kk * 256;

        v2f b;
        b = *(const v2f*)(Bk);
        c0 = __builtin_amdgcn_wmma_f32_16x16x4_f32(false, a, false, b, (short)0, c0, false, false);
        b = *(const v2f*)(Bk + 64);
        c1 = __builtin_amdgcn_wmma_f32_16x16x4_f32(false, a, false, b, (short)0, c1, false, false);
        b = *(const v2f*)(Bk + 128);
        c2 = __builtin_amdgcn_wmma_f32_16x16x4_f32(false, a, false, b, (short)0, c2, false, false);
        b = *(const v2f*)(Bk + 192);
        c3 = __builtin_amdgcn_wmma_f32_16x16x4_f32(false, a, false, b, (short)0, c3, false, false);
    }

    // D layout: VGPR r, lane: M = rowBase + r + hi*8, N = r16 (+16*ct)
#pragma unroll
    for (int r = 0; r < 8; ++r) {
        int m = rowBase + hi * 8 + r;
        float* o = out + (size_t)m * FD;
        float v;
        v = c0[r]; if (BIAS_RELU) { v += bias[     r16]; v = fmaxf(v, 0.0f); } o[     r16] = v;
        v = c1[r]; if (BIAS_RELU) { v += bias[16 + r16]; v = fmaxf(v, 0.0f); } o[16 + r16] = v;
        v = c2[r]; if (BIAS_RELU) { v += bias[32 + r16]; v = fmaxf(v, 0.0f); } o[32 + r16] = v;
        v = c3[r]; if (BIAS_RELU) { v += bias[48 + r16]; v = fmaxf(v, 0.0f); } o[48 + r16] = v;
    }
}

// ---------------------------------------------------------------------------
// Aggregation kernels
// ---------------------------------------------------------------------------
__global__ void k_agg_init(const float* __restrict__ t, const float* __restrict__ nself,
                           float* __restrict__ agg, int N) {
    int g = blockIdx.x * blockDim.x + threadIdx.x;
    if (g >= N * FD) return;
    agg[g] = t[g] * nself[g >> 6];
}

__global__ void k_edge_scatter(const int* __restrict__ src, const int* __restrict__ dst,
                               const float* __restrict__ ne, const float* __restrict__ t,
                               float* __restrict__ agg, int E) {
    int g = blockIdx.x * blockDim.x + threadIdx.x;
    int e = g >> 6;
    if (e >= E) return;
    float w = ne[e];
    if (w == 0.0f) return;
    int f = g & 63;
    atomicAdd(&agg[(size_t)dst[e] * FD + f], t[(size_t)src[e] * FD + f] * w);
}

__global__ void k_bias_relu(float* __restrict__ h, const float* __restrict__ b, int N) {
    int g = blockIdx.x * blockDim.x + threadIdx.x;
    if (g >= N * FD) return;
    h[g] = fmaxf(h[g] + b[g & 63], 0.0f);
}

// ---------------------------------------------------------------------------
// Pooling + head
// ---------------------------------------------------------------------------
__global__ void k_fill0(float* __restrict__ p, int n) {
    int g = blockIdx.x * blockDim.x + threadIdx.x;
    if (g < n) p[g] = 0.0f;
}

__global__ void k_pool(const float* __restrict__ h, const int* __restrict__ batch,
                       const unsigned char* __restrict__ mask, float* __restrict__ pooled, int N) {
    int g = blockIdx.x * blockDim.x + threadIdx.x;
    int n = g >> 6;
    if (n >= N) return;
    if (!mask[n]) return;
    int f = g & 63;
    // post-ReLU values are >= 0; uint compare == float compare; empty graph -> 0 (matches ref)
    atomicMax((unsigned int*)&pooled[(size_t)batch[n] * FD + f],
              __float_as_uint(h[(size_t)n * FD + f]));
}

__global__ void k_head2(const float* __restrict__ h1, const float* __restrict__ fw2,
                        const float* __restrict__ fb2, float* __restrict__ out) {
    int g = blockIdx.x * blockDim.x + threadIdx.x;
    if (g >= GGR * NCLS) return;
    int gi = g / NCLS, c = g % NCLS;
    float acc = fb2[c];
#pragma unroll
    for (int k = 0; k < FD; ++k) acc += h1[gi * FD + k] * fw2[k * NCLS + c];
    out[g] = acc;
}

// ---------------------------------------------------------------------------
// Host side
// ---------------------------------------------------------------------------
extern "C" void kernel_launch(void* const* d_in, const int* in_sizes, int n_in,
                              void* d_out, int out_size, void* d_ws, size_t ws_size,
                              hipStream_t stream) {
    const int N = in_sizes[0] / FD;
    const int E = in_sizes[1] / 2;

    const float*         x     = (const float*)d_in[0];
    const int*           ei    = (const int*)d_in[1];
    const int*           batch = (const int*)d_in[2];
    const unsigned char* mask  = (const unsigned char*)d_in[3];   // jnp bool -> 1 byte/elem
    const float* W1  = (const float*)d_in[4];
    const float* b1  = (const float*)d_in[5];
    const float* W2  = (const float*)d_in[6];
    const float* b2  = (const float*)d_in[7];
    const float* W3  = (const float*)d_in[8];
    const float* b3  = (const float*)d_in[9];
    const float* fw1 = (const float*)d_in[10];
    const float* fb1 = (const float*)d_in[11];
    const float* fw2 = (const float*)d_in[12];
    const float* fb2 = (const float*)d_in[13];
    float* out = (float*)d_out;

    const int* src = ei;
    const int* dst = ei + E;

    // workspace layout (floats)
    float* ws     = (float*)d_ws;
    float* buf0   = ws;                          // N*64
    float* buf1   = buf0 + (size_t)N * FD;       // N*64
    float* tbuf   = buf1 + (size_t)N * FD;       // N*64
    float* ne     = tbuf + (size_t)N * FD;       // E
    float* dinv   = ne + E;                      // N (deg -> dinv in place)
    float* nself  = dinv + N;                    // N
    float* pooled = nself + N;                   // G*64
    float* head1  = pooled + (size_t)GGR * FD;   // G*64

    const int T = 256;
    auto cdiv = [](int a, int b) { return (a + b - 1) / b; };

    // normalization prep
    k_mask_x    <<<cdiv(N * FD, T), T, 0, stream>>>(x, mask, buf0, N);
    k_deg_init  <<<cdiv(N, T),      T, 0, stream>>>(mask, dinv, N);
    k_edge_deg  <<<cdiv(E, T),      T, 0, stream>>>(src, dst, mask, dinv, E);
    k_dinv_nself<<<cdiv(N, T),      T, 0, stream>>>(dinv, nself, N);
    k_norm_e    <<<cdiv(E, T),      T, 0, stream>>>(src, dst, mask, dinv, ne, E);

    // three GCN layers (ping-pong buf0 <-> buf1)
    const float* Ws[3] = {W1, W2, W3};
    const float* bs[3] = {b1, b2, b3};
    float* hin  = buf0;
    float* hagg = buf1;
    const int gemmBlocks = cdiv(cdiv(N, 16), 8);
    for (int l = 0; l < 3; ++l) {
        k_gemm64<false><<<gemmBlocks, T, 0, stream>>>(hin, Ws[l], nullptr, tbuf, N);
        k_agg_init     <<<cdiv(N * FD, T), T, 0, stream>>>(tbuf, nself, hagg, N);
        k_edge_scatter <<<cdiv(E * FD, T), T, 0, stream>>>(src, dst, ne, tbuf, hagg, E);
        k_bias_relu    <<<cdiv(N * FD, T), T, 0, stream>>>(hagg, bs[l], N);
        float* tmp = hin; hin = hagg; hagg = tmp;   // next input = this output
    }
    float* hfinal = hin;   // points at the last layer's output after 3 swaps

    // masked global max-pool
    k_fill0<<<cdiv(GGR * FD, T), T, 0, stream>>>(pooled, GGR * FD);
    k_pool <<<cdiv(N * FD, T),   T, 0, stream>>>(hfinal, batch, mask, pooled, N);

    // MLP head: relu(pooled @ fw1 + fb1) @ fw2 + fb2
    k_gemm64<true><<<cdiv(cdiv(GGR, 16), 8), T, 0, stream>>>(pooled, fw1, fb1, head1, GGR);
    k_head2       <<<cdiv(GGR * NCLS, T),    T, 0, stream>>>(head1, fw2, fb2, out);
}